// AECS_36936718745648
// MI455X (gfx1250) — compile-verified
//
#include <hip/hip_runtime.h>
#include <math.h>

typedef float v2f __attribute__((ext_vector_type(2)));
typedef float v8f __attribute__((ext_vector_type(8)));

#define BB 64
#define TT 512
#define FF 128
#define HH 256
#define LL 64
#define KNN 10
#define FOURH 1024
#define EPSV 1e-8f

// ---------------------------------------------------------------------------
// fp32 WMMA wrapper: D = A(16x4) * B(4x16) + C, all lanes of a wave32.
// ---------------------------------------------------------------------------
static __device__ __forceinline__ v8f wmma4(v2f a, v2f b, v8f c) {
  return __builtin_amdgcn_wmma_f32_16x16x4_f32(false, a, false, b, (short)0, c,
                                               false, false);
}
static __device__ __forceinline__ float sigm(float x) {
  return 1.0f / (1.0f + __expf(-x));
}

// ---------------------------------------------------------------------------
// Tensor Data Mover: DMA a 64x64 fp32 tile (row stride = ld elements) from
// global into LDS with hardware padding of 4 DWORDs every 64 DWORDs, i.e. the
// [64][68] bank-conflict-free layout, tracked by TENSORcnt.  Guarded: falls
// back to cooperative VALU staging when the builtin is unavailable.
// ---------------------------------------------------------------------------
#if defined(__has_builtin)
#if __has_builtin(__builtin_amdgcn_tensor_load_to_lds) && \
    __has_builtin(__builtin_amdgcn_s_wait_tensorcnt)
#define USE_TDM 1
#endif
#endif

#ifdef USE_TDM
typedef unsigned int u32x4 __attribute__((ext_vector_type(4)));
typedef int i32x4 __attribute__((ext_vector_type(4)));
typedef int i32x8 __attribute__((ext_vector_type(8)));

static __device__ __forceinline__ uint32_t lds_byte_off(const void* p) {
  return (uint32_t)(uintptr_t)(__attribute__((address_space(3))) const void*)p;
}

static __device__ __forceinline__ void tdm_load_tile64(const float* gsrc,
                                                       uint32_t ldsoff,
                                                       uint32_t ld_elems) {
  const unsigned long long ga = (unsigned long long)(uintptr_t)gsrc;
  u32x4 g0;
  g0[0] = 1u;                                  // count=1 (valid user D#)
  g0[1] = ldsoff;                              // lds_addr (bytes)
  g0[2] = (uint32_t)(ga & 0xffffffffu);        // global_addr[31:0]
  g0[3] = (uint32_t)((ga >> 32) & 0x01ffffffu) // global_addr[56:32]
          | 0x80000000u;                       // type=2 ("image") bits 127:126
  i32x8 g1;
  g1[0] = (int)((2u << 16)    // data_size = 4 bytes
                | (1u << 20)  // pad_enable
                | (5u << 22)  // pad_interval: 64 DWORDs
                | (3u << 25)); // pad_amount: 4 DWORDs
  g1[1] = (int)(64u << 16);   // tensor_dim0 = 64 (bits 79:48, low half)
  g1[2] = (int)(64u << 16);   // tensor_dim0 hi=0 | tensor_dim1 = 64 (lo16)
  g1[3] = (int)(64u << 16);   // tensor_dim1 hi=0 | tile_dim0 = 64
  g1[4] = (int)64u;           // tile_dim1 = 64, tile_dim2 = 0
  g1[5] = (int)ld_elems;      // tensor_dim0_stride[31:0]
  g1[6] = 0;                  // stride hi / tensor_dim1_stride lo
  g1[7] = 0;                  // tensor_dim1_stride hi
  i32x4 z4;
  z4[0] = 0; z4[1] = 0; z4[2] = 0; z4[3] = 0;
#if __clang_major__ >= 23
  i32x8 z8;
  for (int i = 0; i < 8; ++i) z8[i] = 0;
  __builtin_amdgcn_tensor_load_to_lds(g0, g1, z4, z4, z8, 0);
#else
  __builtin_amdgcn_tensor_load_to_lds(g0, g1, z4, z4, 0);
#endif
}
#endif  // USE_TDM

// ---------------------------------------------------------------------------
// Fused masked-pdist: one pass computes cnt, V2m.M, M.V2m, Vm.Vm with WMMA
// and writes the distance matrix (inf on diagonal / cnt<=0).
// View of x/mask selected by (rowStride,colStride): space=(F,1), time=(1,F).
// ---------------------------------------------------------------------------
__global__ __launch_bounds__(256) void pdist_kernel(
    const float* __restrict__ x, const float* __restrict__ mask,
    float* __restrict__ dout, int Nrows, int Kdim, int rowStride,
    int colStride) {
  __shared__ float Mn[64][36], Vn[64][36], Wn[64][36];
  __shared__ float Mm[64][36], Vm_[64][36], Wm[64][36];
  const int tid = threadIdx.x, wave = tid >> 5, lane = tid & 31;
  const int lr = lane & 15, lh = lane >> 4;
  const int row0 = blockIdx.x * 64, col0 = blockIdx.y * 64;
  const size_t bbase = (size_t)blockIdx.z * (size_t)(TT * FF);

  v8f acc[2][4];
  for (int s = 0; s < 2; ++s)
    for (int q = 0; q < 4; ++q)
      for (int i = 0; i < 8; ++i) acc[s][q][i] = 0.f;

  for (int k0 = 0; k0 < Kdim; k0 += 32) {
    for (int i = tid; i < 64 * 32; i += 256) {
      int r = i >> 5, k = i & 31;
      size_t go = bbase + (size_t)(row0 + r) * rowStride +
                  (size_t)(k0 + k) * colStride;
      float m = mask[go], v = x[go];
      Mn[r][k] = m; Vn[r][k] = v * m; Wn[r][k] = v * v * m;
      go = bbase + (size_t)(col0 + r) * rowStride +
           (size_t)(k0 + k) * colStride;
      m = mask[go]; v = x[go];
      Mm[r][k] = m; Vm_[r][k] = v * m; Wm[r][k] = v * v * m;
    }
    __syncthreads();
    for (int s = 0; s < 2; ++s) {
      const int si = wave * 2 + s, mi = si >> 2, ni = si & 3;
      const int ar = mi * 16 + lr, bc = ni * 16 + lr;
      for (int kk = 0; kk < 32; kk += 4) {
        const int kb = kk + lh * 2;
        v2f aM, aV, aW, bM, bV, bW;
        aM[0] = Mn[ar][kb];  aM[1] = Mn[ar][kb + 1];
        aV[0] = Vn[ar][kb];  aV[1] = Vn[ar][kb + 1];
        aW[0] = Wn[ar][kb];  aW[1] = Wn[ar][kb + 1];
        bM[0] = Mm[bc][kb];  bM[1] = Mm[bc][kb + 1];
        bV[0] = Vm_[bc][kb]; bV[1] = Vm_[bc][kb + 1];
        bW[0] = Wm[bc][kb];  bW[1] = Wm[bc][kb + 1];
        acc[s][0] = wmma4(aM, bM, acc[s][0]);  // cnt
        acc[s][1] = wmma4(aW, bM, acc[s][1]);  // V2m_n . M_m
        acc[s][2] = wmma4(aM, bW, acc[s][2]);  // M_n . V2m_m
        acc[s][3] = wmma4(aV, bV, acc[s][3]);  // Vm . Vm
      }
    }
    __syncthreads();
  }
  const size_t dbase = (size_t)blockIdx.z * (size_t)Nrows * (size_t)Nrows;
  for (int s = 0; s < 2; ++s) {
    const int si = wave * 2 + s, mi = si >> 2, ni = si & 3;
    const int col = col0 + ni * 16 + lr;
    for (int v = 0; v < 8; ++v) {
      const int row = row0 + mi * 16 + v + lh * 8;
      float cnt = acc[s][0][v];
      float sq = acc[s][1][v] + acc[s][2][v] - 2.0f * acc[s][3][v];
      sq = fmaxf(sq, 0.0f);
      const bool invalid = (cnt <= 0.0f) || (row == col);
      float d = invalid ? __builtin_inff() : sqrtf(sq / (cnt + EPSV));
      dout[dbase + (size_t)row * Nrows + col] = d;
    }
  }
}

// ---------------------------------------------------------------------------
// Top-K(10) smallest per row + gaussian weights. One wave per row.
// ---------------------------------------------------------------------------
__global__ __launch_bounds__(32) void topk_kernel(
    const float* __restrict__ dmat, int N, int* __restrict__ idx_out,
    float* __restrict__ w_out) {
  __shared__ float buf[512];
  const int row = blockIdx.x;
  const int lane = threadIdx.x;
  const float* dr = dmat + (size_t)row * N;
  for (int j = lane; j < N; j += 32) buf[j] = dr[j];
  __syncthreads();
  float td[KNN];
  int ti[KNN];
  for (int k = 0; k < KNN; ++k) {
    float best = __builtin_inff();
    int bi = 0x7fffffff;
    for (int j = lane; j < N; j += 32) {
      float v = buf[j];
      if (v < best || (v == best && j < bi)) { best = v; bi = j; }
    }
    for (int off = 16; off; off >>= 1) {
      float ov = __shfl_xor(best, off, 32);
      int oi = __shfl_xor(bi, off, 32);
      if (ov < best || (ov == best && oi < bi)) { best = ov; bi = oi; }
    }
    td[k] = best;
    ti[k] = bi;
    if (lane == 0 && bi >= 0 && bi < N) buf[bi] = __builtin_inff();
    __syncthreads();
  }
  if (lane == 0) {
    float vc = 0.f, ssum = 0.f, safe[KNN];
    for (int k = 0; k < KNN; ++k) {
      const bool valid = td[k] < 3.0e38f;
      safe[k] = valid ? td[k] : 0.f;
      vc += valid ? 1.f : 0.f;
      ssum += safe[k];
    }
    float sigma = ssum / (vc + EPSV);
    if (!(vc > 0.f)) sigma = 1.f;
    const float s2 = sigma * sigma + EPSV;
    float w[KNN], wsum = 0.f;
    for (int k = 0; k < KNN; ++k) {
      const bool valid = td[k] < 3.0e38f;
      w[k] = valid ? __expf(-(safe[k] * safe[k]) / s2) : 0.f;
      wsum += w[k];
    }
    for (int k = 0; k < KNN; ++k) {
      idx_out[(size_t)row * KNN + k] = ti[k];
      w_out[(size_t)row * KNN + k] = w[k] / (wsum + EPSV);
    }
  }
}

// ---------------------------------------------------------------------------
// kNN fill along T (space) / along F (time).
// ---------------------------------------------------------------------------
__global__ void fill_space_kernel(const float* __restrict__ x,
                                  const float* __restrict__ mask,
                                  const int* __restrict__ idx,
                                  const float* __restrict__ w,
                                  float* __restrict__ out) {
  const int i = blockIdx.x * blockDim.x + threadIdx.x;
  if (i >= BB * TT * FF) return;
  const int f = i % FF;
  const int t = (i / FF) % TT;
  const int b = i / (FF * TT);
  const float m = mask[i], v = x[i];
  if (m > 0.f) { out[i] = v; return; }
  const int row = b * TT + t;
  float ws = 0.f, acc = 0.f;
  for (int k = 0; k < KNN; ++k) {
    const int id = idx[(size_t)row * KNN + k];
    const float wk = w[(size_t)row * KNN + k];
    const size_t off = ((size_t)b * TT + id) * FF + f;
    const float no = mask[off];
    acc += x[off] * no * wk;
    ws += no * wk;
  }
  out[i] = (ws > 0.f) ? acc / (ws + EPSV) : 0.f;
}

__global__ void fill_time_kernel(const float* __restrict__ x,
                                 const float* __restrict__ mask,
                                 const int* __restrict__ idx,
                                 const float* __restrict__ w,
                                 float* __restrict__ out) {
  const int i = blockIdx.x * blockDim.x + threadIdx.x;
  if (i >= BB * TT * FF) return;
  const int f = i % FF;
  const int t = (i / FF) % TT;
  const int b = i / (FF * TT);
  const float m = mask[i], v = x[i];
  if (m > 0.f) { out[i] = v; return; }
  const int row = b * FF + f;
  float ws = 0.f, acc = 0.f;
  for (int k = 0; k < KNN; ++k) {
    const int id = idx[(size_t)row * KNN + k];
    const float wk = w[(size_t)row * KNN + k];
    const size_t off = ((size_t)b * TT + t) * FF + id;
    const float no = mask[off];
    acc += x[off] * no * wk;
    ws += no * wk;
  }
  out[i] = (ws > 0.f) ? acc / (ws + EPSV) : 0.f;
}

// ---------------------------------------------------------------------------
// Pack [x_in, mask] -> (B*T, 2F). mulmask=1 multiplies x by mask (x_zero).
// ---------------------------------------------------------------------------
__global__ void pack_kernel(const float* __restrict__ xin,
                            const float* __restrict__ mask,
                            float* __restrict__ P, int mulmask) {
  const int i = blockIdx.x * blockDim.x + threadIdx.x;
  if (i >= BB * TT * FF) return;
  const int f = i % FF;
  const size_t row = i / FF;  // b*T + t
  const float m = mask[i];
  float v = xin[i];
  if (mulmask) v *= m;
  P[row * (2 * FF) + f] = v;
  P[row * (2 * FF) + FF + f] = m;
}

// ---------------------------------------------------------------------------
// Generic fp32 WMMA GEMM: C(rows,N) = A(rows,K) @ W(K,N) + bias.
// 256 threads (8 waves) -> 64x64 tile, 2 subtiles/wave.  Tile staging uses
// the Tensor Data Mover (double-buffered, TENSORcnt split-wait, hardware LDS
// padding to [64][68]) when available; VALU staging otherwise.
// Dynamic LDS: 2 * (As + Bs) * 64*68 floats = 69632 bytes.
// ---------------------------------------------------------------------------
#define GEMM_LDS_FLOATS (4 * 64 * 68)
__global__ __launch_bounds__(256) void gemm_bias_kernel(
    const float* __restrict__ A, const float* __restrict__ W,
    const float* __restrict__ bias, float* __restrict__ C, int rows, int Kdim,
    int Ndim) {
  extern __shared__ float smem[];
  float* As0 = smem;                  // [2][64][68]
  float* Bs0 = smem + 2 * 64 * 68;    // [2][64][68]
  const int tid = threadIdx.x, wave = tid >> 5, lane = tid & 31;
  const int lr = lane & 15, lh = lane >> 4;
  const int row0 = blockIdx.x * 64, col0 = blockIdx.y * 64;
  const int nchunks = Kdim >> 6;

  v8f acc[2];
  for (int s = 0; s < 2; ++s)
    for (int i = 0; i < 8; ++i) acc[s][i] = 0.f;

#ifdef USE_TDM
  const bool issuer = (wave == 0);
  if (issuer) {
    tdm_load_tile64(A + (size_t)row0 * Kdim, lds_byte_off(As0), Kdim);
    tdm_load_tile64(W + col0, lds_byte_off(Bs0), Ndim);
  }
#endif

  for (int ci = 0; ci < nchunks; ++ci) {
    const int k0 = ci << 6;
    const int buf = ci & 1;
    float* Asb = As0 + buf * 64 * 68;
    float* Bsb = Bs0 + buf * 64 * 68;
#ifdef USE_TDM
    if (issuer) {
      if (ci + 1 < nchunks) {
        const int kn = (ci + 1) << 6;
        float* Asn = As0 + ((ci + 1) & 1) * 64 * 68;
        float* Bsn = Bs0 + ((ci + 1) & 1) * 64 * 68;
        tdm_load_tile64(A + (size_t)row0 * Kdim + kn, lds_byte_off(Asn), Kdim);
        tdm_load_tile64(W + (size_t)kn * Ndim + col0, lds_byte_off(Bsn), Ndim);
        __builtin_amdgcn_s_wait_tensorcnt(2);  // current pair landed
      } else {
        __builtin_amdgcn_s_wait_tensorcnt(0);
      }
    }
    __syncthreads();
#else
    for (int i = tid; i < 64 * 16; i += 256) {
      const int r = i >> 4, c4 = (i & 15) << 2;
      const float4 va =
          *(const float4*)(A + (size_t)(row0 + r) * Kdim + k0 + c4);
      Asb[r * 68 + c4 + 0] = va.x; Asb[r * 68 + c4 + 1] = va.y;
      Asb[r * 68 + c4 + 2] = va.z; Asb[r * 68 + c4 + 3] = va.w;
      const float4 vb =
          *(const float4*)(W + (size_t)(k0 + r) * Ndim + col0 + c4);
      Bsb[r * 68 + c4 + 0] = vb.x; Bsb[r * 68 + c4 + 1] = vb.y;
      Bsb[r * 68 + c4 + 2] = vb.z; Bsb[r * 68 + c4 + 3] = vb.w;
    }
    __syncthreads();
#endif
    for (int s = 0; s < 2; ++s) {
      const int si = wave * 2 + s, mi = si >> 2, ni = si & 3;
      const int ar = mi * 16 + lr, bc = ni * 16 + lr;
      v8f a8 = acc[s];
      for (int kk = 0; kk < 64; kk += 4) {
        const int kb = kk + lh * 2;
        v2f a, b;
        a[0] = Asb[ar * 68 + kb]; a[1] = Asb[ar * 68 + kb + 1];
        b[0] = Bsb[kb * 68 + bc]; b[1] = Bsb[(kb + 1) * 68 + bc];
        a8 = wmma4(a, b, a8);
      }
      acc[s] = a8;
    }
    __syncthreads();
  }
  for (int s = 0; s < 2; ++s) {
    const int si = wave * 2 + s, mi = si >> 2, ni = si & 3;
    const int col = col0 + ni * 16 + lr;
    const float bv = bias ? bias[col] : 0.f;
    for (int v = 0; v < 8; ++v) {
      const int r = row0 + mi * 16 + v + lh * 8;
      C[(size_t)r * Ndim + col] = acc[s][v] + bv;
    }
  }
}

// ---------------------------------------------------------------------------
// Persistent LSTM recurrence. One workgroup (512 threads = 16 waves).
// XW = x@W + b precomputed (B*T,1024) with row = b*T + t.
// h kept in LDS ping-pong (2 x 64 x 260 floats), c entirely in VGPRs.
// Each wave owns 4 (m-tile, j-tile) groups; a group computes all 4 gate
// tiles (cols j, j+256, j+512, j+768) so the gate combine is wave-local.
// ---------------------------------------------------------------------------
__global__ __launch_bounds__(512) void lstm_kernel(
    const float* __restrict__ XW, const float* __restrict__ U,
    float* __restrict__ HS) {
  extern __shared__ float hbuf[];
  const int HP = HH + 4;  // 260, bank-conflict-free column reads
  const int tid = threadIdx.x;
  const int wave = tid >> 5, lane = tid & 31;
  const int lr = lane & 15, lh = lane >> 4;

  v8f creg[4];
  for (int g = 0; g < 4; ++g)
    for (int i = 0; i < 8; ++i) creg[g][i] = 0.f;
  for (int i = tid; i < 2 * 64 * HP; i += 512) hbuf[i] = 0.f;
  __syncthreads();

  for (int t = 0; t < TT; ++t) {
    const float* hc = hbuf + (size_t)(t & 1) * 64 * HP;
    float* hn = hbuf + (size_t)((t + 1) & 1) * 64 * HP;
    for (int g = 0; g < 4; ++g) {
      const int gid = wave * 4 + g;
      const int mi = gid >> 4, ji = gid & 15;
      const int ar = mi * 16 + lr;
      const int colb = ji * 16 + lr;
      v8f z0, z1, z2, z3;
      for (int i = 0; i < 8; ++i) { z0[i] = 0.f; z1[i] = 0.f; z2[i] = 0.f; z3[i] = 0.f; }
      for (int kk = 0; kk < HH; kk += 4) {
        const int kb = kk + lh * 2;
        v2f a, b;
        a[0] = hc[ar * HP + kb];
        a[1] = hc[ar * HP + kb + 1];
        b[0] = U[(size_t)kb * FOURH + colb];
        b[1] = U[(size_t)(kb + 1) * FOURH + colb];
        z0 = wmma4(a, b, z0);
        b[0] = U[(size_t)kb * FOURH + 256 + colb];
        b[1] = U[(size_t)(kb + 1) * FOURH + 256 + colb];
        z1 = wmma4(a, b, z1);
        b[0] = U[(size_t)kb * FOURH + 512 + colb];
        b[1] = U[(size_t)(kb + 1) * FOURH + 512 + colb];
        z2 = wmma4(a, b, z2);
        b[0] = U[(size_t)kb * FOURH + 768 + colb];
        b[1] = U[(size_t)(kb + 1) * FOURH + 768 + colb];
        z3 = wmma4(a, b, z3);
      }
      for (int v = 0; v < 8; ++v) {
        const int brow = mi * 16 + v + lh * 8;
        const int j = ji * 16 + lr;
        const size_t xo = ((size_t)brow * TT + t) * FOURH;
        const float zi = z0[v] + XW[xo + j];
        const float zf = z1[v] + XW[xo + 256 + j];
        const float zg = z2[v] + XW[xo + 512 + j];
        const float zo = z3[v] + XW[xo + 768 + j];
        const float c = sigm(zf) * creg[g][v] + sigm(zi) * tanhf(zg);
        creg[g][v] = c;
        const float h = sigm(zo) * tanhf(c);
        hn[brow * HP + j] = h;
        HS[((size_t)brow * TT + t) * HH + j] = h;
      }
    }
    __syncthreads();
  }
}

// ---------------------------------------------------------------------------
// LayerNorm over 256 features, one wave per row, in-place.
// ---------------------------------------------------------------------------
__global__ __launch_bounds__(256) void ln_kernel(float* __restrict__ hs,
                                                 const float* __restrict__ g,
                                                 const float* __restrict__ b,
                                                 int rows) {
  const int row = blockIdx.x * 8 + (threadIdx.x >> 5);
  const int lane = threadIdx.x & 31;
  if (row >= rows) return;
  float* p = hs + (size_t)row * HH;
  float vals[8], s = 0.f;
  for (int i = 0; i < 8; ++i) { vals[i] = p[lane + i * 32]; s += vals[i]; }
  for (int off = 16; off; off >>= 1) s += __shfl_xor(s, off, 32);
  const float mean = s * (1.f / 256.f);
  float vs = 0.f;
  for (int i = 0; i < 8; ++i) { const float d = vals[i] - mean; vs += d * d; }
  for (int off = 16; off; off >>= 1) vs += __shfl_xor(vs, off, 32);
  const float inv = 1.0f / sqrtf(vs * (1.f / 256.f) + 1e-3f);
  for (int i = 0; i < 8; ++i) {
    const int j = lane + i * 32;
    p[j] = (vals[i] - mean) * inv * g[j] + b[j];
  }
}

// ---------------------------------------------------------------------------
// L2 normalize rows of (rows, 64), one wave per row, in-place.
// ---------------------------------------------------------------------------
__global__ __launch_bounds__(256) void l2n_kernel(float* __restrict__ z,
                                                  int rows) {
  const int row = blockIdx.x * 8 + (threadIdx.x >> 5);
  const int lane = threadIdx.x & 31;
  if (row >= rows) return;
  float* p = z + (size_t)row * LL;
  const float a = p[lane], c = p[lane + 32];
  float ss = a * a + c * c;
  for (int off = 16; off; off >>= 1) ss += __shfl_xor(ss, off, 32);
  const float inv = 1.0f / sqrtf(fmaxf(ss, 1e-12f));
  p[lane] = a * inv;
  p[lane + 32] = c * inv;
}

// ---------------------------------------------------------------------------
// Mean-pool z over T into pooled(B,193) cols 0..191; missing rate -> col 192.
// ---------------------------------------------------------------------------
__global__ void pool_kernel(const float* __restrict__ zo,
                            const float* __restrict__ zs,
                            const float* __restrict__ zt,
                            float* __restrict__ pooled) {
  const int i = blockIdx.x * blockDim.x + threadIdx.x;
  if (i >= BB * 192) return;
  const int b = i / 192, c = i % 192;
  const float* z = (c < 64) ? zo : (c < 128) ? zs : zt;
  const int l = c & 63;
  float s = 0.f;
  for (int t = 0; t < TT; ++t) s += z[((size_t)b * TT + t) * LL + l];
  pooled[b * 193 + c] = s * (1.0f / TT);
}

__global__ __launch_bounds__(256) void missing_kernel(
    const float* __restrict__ mask, float* __restrict__ pooled) {
  __shared__ float red[256];
  const int b = blockIdx.x;
  float s = 0.f;
  for (int i = threadIdx.x; i < TT * FF; i += 256)
    s += mask[(size_t)b * TT * FF + i];
  red[threadIdx.x] = s;
  __syncthreads();
  for (int st = 128; st; st >>= 1) {
    if (threadIdx.x < st) red[threadIdx.x] += red[threadIdx.x + st];
    __syncthreads();
  }
  if (threadIdx.x == 0)
    pooled[b * 193 + 192] = 1.f - red[0] / (float)(TT * FF);
}

// ---------------------------------------------------------------------------
// Gating MLP: 193 -> 128 -> 64 -> softmax(3). One block per batch element.
// ---------------------------------------------------------------------------
__global__ __launch_bounds__(128) void gate_kernel(
    const float* __restrict__ pooled, const float* __restrict__ W1,
    const float* __restrict__ b1, const float* __restrict__ W2,
    const float* __restrict__ b2, const float* __restrict__ Wa,
    const float* __restrict__ ba, float* __restrict__ alpha) {
  __shared__ float pr[193];
  __shared__ float h1[128];
  __shared__ float h2[64];
  const int b = blockIdx.x, tid = threadIdx.x;
  for (int i = tid; i < 193; i += 128) pr[i] = pooled[b * 193 + i];
  __syncthreads();
  if (tid < 128) {
    float s = b1[tid];
    for (int k = 0; k < 193; ++k) s += pr[k] * W1[k * 128 + tid];
    h1[tid] = 1.f - __expf(-s * s);
  }
  __syncthreads();
  if (tid < 64) {
    float s = b2[tid];
    for (int k = 0; k < 128; ++k) s += h1[k] * W2[k * 64 + tid];
    h2[tid] = 1.f - __expf(-s * s);
  }
  __syncthreads();
  if (tid == 0) {
    float lg[3], mx = -1e30f;
    for (int j = 0; j < 3; ++j) {
      float s = ba[j];
      for (int k = 0; k < 64; ++k) s += h2[k] * Wa[k * 3 + j];
      lg[j] = s;
      mx = fmaxf(mx, s);
    }
    float se = 0.f;
    for (int j = 0; j < 3; ++j) { lg[j] = __expf(lg[j] - mx); se += lg[j]; }
    for (int j = 0; j < 3; ++j) alpha[b * 3 + j] = lg[j] / se;
  }
}

__global__ void fuse_kernel(const float* __restrict__ zo,
                            const float* __restrict__ zs,
                            const float* __restrict__ zt,
                            const float* __restrict__ alpha,
                            float* __restrict__ zf) {
  const int i = blockIdx.x * blockDim.x + threadIdx.x;
  if (i >= BB * TT * LL) return;
  const int b = i / (TT * LL);
  zf[i] = alpha[b * 3 + 0] * zo[i] + alpha[b * 3 + 1] * zs[i] +
          alpha[b * 3 + 2] * zt[i];
}

__global__ void combine_kernel(const float* __restrict__ x,
                               const float* __restrict__ mask,
                               const float* __restrict__ xh,
                               float* __restrict__ out) {
  const int i = blockIdx.x * blockDim.x + threadIdx.x;
  if (i >= BB * TT * FF) return;
  const float m = mask[i];
  const float h = fminf(fmaxf(xh[i], -5.f), 5.f);
  out[i] = m * x[i] + (1.f - m) * h;
}

// ---------------------------------------------------------------------------
// Host driver
// ---------------------------------------------------------------------------
extern "C" void kernel_launch(void* const* d_in, const int* in_sizes, int n_in,
                              void* d_out, int out_size, void* d_ws,
                              size_t ws_size, hipStream_t stream) {
  (void)in_sizes; (void)n_in; (void)out_size; (void)ws_size;
  auto in = [&](int i) { return (const float*)d_in[i]; };
  const float* X = in(0);
  const float* MASK = in(1);

  struct EncP {
    const float *W1, *U1, *b1, *W2, *U2, *b2, *g1, *bb1, *g2, *bb2, *Wl, *bl;
  };
  EncP enc[3];
  for (int e = 0; e < 3; ++e) {
    const int p = 2 + e * 12;
    enc[e] = {in(p + 0), in(p + 1), in(p + 2), in(p + 3), in(p + 4), in(p + 5),
              in(p + 6), in(p + 7), in(p + 8), in(p + 9), in(p + 10), in(p + 11)};
  }
  EncP dec = {in(38), in(39), in(40), in(41), in(42), in(43),
              in(44), in(45), in(46), in(47), in(48), in(49)};
  const float *gW1 = in(50), *gb1 = in(51), *gW2 = in(52), *gb2 = in(53),
              *gWa = in(54), *gba = in(55);

  // Workspace layout (floats), 256B aligned chunks.
  float* w0 = (float*)d_ws;
  size_t off = 0;
  auto alloc = [&](size_t n) {
    float* p = w0 + off;
    off += (n + 63) & ~(size_t)63;
    return p;
  };
  float* P = alloc((size_t)BB * TT * 2 * FF);        // 8.4M  (pack buffer)
  float* XW = alloc((size_t)BB * TT * FOURH);        // 33.6M (gate pre-acts)
  float* HS1 = alloc((size_t)BB * TT * HH);          // 8.4M
  float* HS2 = alloc((size_t)BB * TT * HH);          // 8.4M
  float* ZO = alloc((size_t)BB * TT * LL);
  float* ZS = alloc((size_t)BB * TT * LL);
  float* ZT = alloc((size_t)BB * TT * LL);
  float* ZF = alloc((size_t)BB * TT * LL);
  float* XSP = alloc((size_t)BB * TT * FF);
  float* XTI = alloc((size_t)BB * TT * FF);
  int* IDXS = (int*)alloc((size_t)BB * TT * KNN);
  float* WSP = alloc((size_t)BB * TT * KNN);
  int* IDXT = (int*)alloc((size_t)BB * FF * KNN);
  float* WTI = alloc((size_t)BB * FF * KNN);
  float* POOLED = alloc(BB * 193);
  float* ALPHA = alloc(BB * 3);
  // Time-disjoint aliases:
  float* DSP = XW;                      // (B,512,512) lives before XW is used
  float* DTI = XW + (size_t)16777216;   // (B,128,128)
  float* XH = P;                        // decoder output proj reuses P

  const int rows = BB * TT;  // 32768
  const int nElem = BB * TT * FF;
  const dim3 blk256(256);
  const int gElem = (nElem + 255) / 256;
  const size_t lstmShm = (size_t)2 * 64 * (HH + 4) * sizeof(float);  // 133 KB
  const size_t gemmShm = (size_t)GEMM_LDS_FLOATS * sizeof(float);    // 69632 B

  // ---- Phase 1: kNN fills ----
  pdist_kernel<<<dim3(8, 8, BB), blk256, 0, stream>>>(X, MASK, DSP, TT, FF, FF, 1);
  topk_kernel<<<BB * TT, 32, 0, stream>>>(DSP, TT, IDXS, WSP);
  fill_space_kernel<<<gElem, blk256, 0, stream>>>(X, MASK, IDXS, WSP, XSP);

  pdist_kernel<<<dim3(2, 2, BB), blk256, 0, stream>>>(X, MASK, DTI, FF, TT, 1, FF);
  topk_kernel<<<BB * FF, 32, 0, stream>>>(DTI, FF, IDXT, WTI);
  fill_time_kernel<<<gElem, blk256, 0, stream>>>(X, MASK, IDXT, WTI, XTI);

  // ---- Phase 2: encoders ----
  auto run_encoder = [&](const float* xin, int mulmask, const EncP& e,
                         float* zout) {
    pack_kernel<<<gElem, blk256, 0, stream>>>(xin, MASK, P, mulmask);
    gemm_bias_kernel<<<dim3(rows / 64, FOURH / 64), blk256, gemmShm, stream>>>(
        P, e.W1, e.b1, XW, rows, 2 * FF, FOURH);
    lstm_kernel<<<1, 512, lstmShm, stream>>>(XW, e.U1, HS1);
    ln_kernel<<<rows / 8, blk256, 0, stream>>>(HS1, e.g1, e.bb1, rows);
    gemm_bias_kernel<<<dim3(rows / 64, FOURH / 64), blk256, gemmShm, stream>>>(
        HS1, e.W2, e.b2, XW, rows, HH, FOURH);
    lstm_kernel<<<1, 512, lstmShm, stream>>>(XW, e.U2, HS2);
    ln_kernel<<<rows / 8, blk256, 0, stream>>>(HS2, e.g2, e.bb2, rows);
    gemm_bias_kernel<<<dim3(rows / 64, LL / 64), blk256, gemmShm, stream>>>(
        HS2, e.Wl, e.bl, zout, rows, HH, LL);
    l2n_kernel<<<rows / 8, blk256, 0, stream>>>(zout, rows);
  };
  run_encoder(X, 1, enc[0], ZO);    // orig: x*mask
  run_encoder(XSP, 0, enc[1], ZS);  // space fill
  run_encoder(XTI, 0, enc[2], ZT);  // time fill

  // ---- Phase 3: gating + fusion ----
  pool_kernel<<<(BB * 192 + 255) / 256, blk256, 0, stream>>>(ZO, ZS, ZT, POOLED);
  missing_kernel<<<BB, blk256, 0, stream>>>(MASK, POOLED);
  gate_kernel<<<BB, 128, 0, stream>>>(POOLED, gW1, gb1, gW2, gb2, gWa, gba, ALPHA);
  fuse_kernel<<<(BB * TT * LL + 255) / 256, blk256, 0, stream>>>(ZO, ZS, ZT, ALPHA, ZF);

  // ---- Phase 4: decoder ----
  gemm_bias_kernel<<<dim3(rows / 64, FOURH / 64), blk256, gemmShm, stream>>>(
      ZF, dec.W1, dec.b1, XW, rows, LL, FOURH);
  lstm_kernel<<<1, 512, lstmShm, stream>>>(XW, dec.U1, HS1);
  ln_kernel<<<rows / 8, blk256, 0, stream>>>(HS1, dec.g1, dec.bb1, rows);
  gemm_bias_kernel<<<dim3(rows / 64, FOURH / 64), blk256, gemmShm, stream>>>(
      HS1, dec.W2, dec.b2, XW, rows, HH, FOURH);
  lstm_kernel<<<1, 512, lstmShm, stream>>>(XW, dec.U2, HS2);
  ln_kernel<<<rows / 8, blk256, 0, stream>>>(HS2, dec.g2, dec.bb2, rows);
  gemm_bias_kernel<<<dim3(rows / 64, FF / 64), blk256, gemmShm, stream>>>(
      HS2, dec.Wl /*dec_Wo*/, dec.bl /*dec_bo*/, XH, rows, HH, FF);
  combine_kernel<<<gElem, blk256, 0, stream>>>(X, MASK, XH, (float*)d_out);
}